// GMMLoss_27633819583115
// MI455X (gfx1250) — compile-verified
//
#include <hip/hip_runtime.h>
#include <math.h>

// ---------------------------------------------------------------------------
// GMM loss (DAGMM-style) for z:[N,8] f32, gamma:[N,4] f32, scalar f32 output.
//
// Pass 1 (gmm_stats_pass): one sweep over z+gamma computing
//   sum_gamma[k], S_z[k][d] = sum gamma*z, S_zz[k][d][e] = sum gamma*z_d*z_e.
//   S_zz runs on the matrix pipe: V_WMMA_F32_16X16X4_F32 rank-4 updates with
//   the asymmetric factorization A = gamma*z, B = z (no sqrt needed),
//   packing 2 mixture components per 16x16 tile (diagonal 8x8 blocks used).
// Pass 2 (gmm_model_build, 1 lane): cov = S_zz/sg - mu mu^T + eps*I,
//   Cholesky, det, inverse, cov_diag, mixture coefficients.
// Pass 3 (gmm_energy): per-sample quadratic forms + -log(sum phi*e/sqrt(det)),
//   block reduce, global atomic.
// Pass 4 (gmm_finalize): scalar combine.
// ---------------------------------------------------------------------------

typedef __attribute__((ext_vector_type(2))) float v2f;
typedef __attribute__((ext_vector_type(8))) float v8f;

#define EPSV 1e-9f

// workspace layout (float offsets)
#define WS_SG       0    // [4]   sum_gamma
#define WS_SZ       4    // [4][8] sum gamma*z
#define WS_SZZ      36   // [4][64] sum gamma*z*z^T
#define WS_ENERGY   292  // scalar accumulator
#define WS_COVDIAG  293  // scalar
#define WS_P        296  // [4][64] cov inverse
#define WS_MU       552  // [4][8]
#define WS_COEF     584  // [4]  phi / sqrt(det_cov)
#define WS_ZERO_FLOATS 296   // accumulators that must start at 0 each launch

#define BLOCKS1   256
#define THREADS1  256
#define WAVES_PER_BLOCK (THREADS1 / 32)

__global__ void __launch_bounds__(THREADS1)
gmm_stats_pass(const float* __restrict__ z, const float* __restrict__ gamma,
               float* __restrict__ ws, int N) {
    __shared__ float sAcc[292];   // mirrors ws[0..291]
    for (int i = threadIdx.x; i < 292; i += blockDim.x) sAcc[i] = 0.f;
    __syncthreads();

    const int lane = threadIdx.x & 31;
    const int wave = threadIdx.x >> 5;
    const int gw   = blockIdx.x * WAVES_PER_BLOCK + wave;
    const int totalWaves = gridDim.x * WAVES_PER_BLOCK;
    const int spw  = N / totalWaves;          // N = 2^19, waves = 2^11 -> 256
    const int base = gw * spw;
    const int iters = spw >> 2;               // 4 samples per WMMA iteration

    // f32 WMMA A-operand layout: lane holds row M = lane&15;
    // VGPR0/1 = K=0/1 (lanes 0-15) or K=2/3 (lanes 16-31). B is symmetric
    // (lane = column N = lane&15), so both operands use the same per-lane
    // (component, dim) mapping.
    const int r  = lane & 15;   // packed row: [0..7]=comp a dims, [8..15]=comp b dims
    const int d  = r & 7;       // feature dim
    const int kh = r >> 3;      // component-within-pair (0/1)
    const int soff = (lane >= 16) ? 2 : 0;    // which 2 of the 4 samples

    v8f c01 = {};               // S_zz tile for components {0,1}
    v8f c23 = {};               // S_zz tile for components {2,3}
    float sz01 = 0.f, sz23 = 0.f, g01 = 0.f, g23 = 0.f;
    const float gsel = (d == 0) ? 1.f : 0.f;  // avoid divergence around WMMA

    for (int i = 0; i < iters; ++i) {
        const int sA = base + 4 * i + soff;
        const int sB = sA + 1;
        const float zA = z[sA * 8 + d];
        const float zB = z[sB * 8 + d];
        const float gA0 = gamma[sA * 4 + kh];
        const float gB0 = gamma[sB * 4 + kh];
        const float gA2 = gamma[sA * 4 + 2 + kh];
        const float gB2 = gamma[sB * 4 + 2 + kh];

        // Asymmetric factorization: C += (gamma*z)^T * z  (no sqrt needed)
        v2f bz;   bz.x  = zA;        bz.y  = zB;        // B: plain z
        v2f a01;  a01.x = gA0 * zA;  a01.y = gB0 * zB;  // A: gamma-weighted z
        v2f a23;  a23.x = gA2 * zA;  a23.y = gB2 * zB;

        c01 = __builtin_amdgcn_wmma_f32_16x16x4_f32(
                  false, a01, false, bz, (short)0, c01, false, false);
        c23 = __builtin_amdgcn_wmma_f32_16x16x4_f32(
                  false, a23, false, bz, (short)0, c23, false, false);

        sz01 += gA0 * zA + gB0 * zB;          // S_z for comp kh
        sz23 += gA2 * zA + gB2 * zB;          // S_z for comp 2+kh
        g01  += gsel * (gA0 + gB0);           // sum_gamma (only d==0 lanes)
        g23  += gsel * (gA2 + gB2);
    }

    // fold this wave's partials into LDS (ds_add_f32)
    atomicAdd(&sAcc[WS_SZ + kh * 8 + d], sz01);
    atomicAdd(&sAcc[WS_SZ + (2 + kh) * 8 + d], sz23);
    if (d == 0) {
        atomicAdd(&sAcc[WS_SG + kh], g01);
        atomicAdd(&sAcc[WS_SG + 2 + kh], g23);
    }
    // C/D layout: VGPR j -> row M=j (lanes 0-15, N=lane) or M=j+8 (lanes 16-31,
    // N=lane-16). Diagonal 8x8 blocks hold the two component matrices.
    if (lane < 8) {                            // comp a block: M=j<8, e=lane<8
        #pragma unroll
        for (int j = 0; j < 8; ++j) {
            atomicAdd(&sAcc[WS_SZZ + 0 * 64 + j * 8 + lane], c01[j]);
            atomicAdd(&sAcc[WS_SZZ + 2 * 64 + j * 8 + lane], c23[j]);
        }
    } else if (lane >= 24) {                   // comp b block: M=j+8, e=lane-24
        const int e = lane - 24;
        #pragma unroll
        for (int j = 0; j < 8; ++j) {
            atomicAdd(&sAcc[WS_SZZ + 1 * 64 + j * 8 + e], c01[j]);
            atomicAdd(&sAcc[WS_SZZ + 3 * 64 + j * 8 + e], c23[j]);
        }
    }
    __syncthreads();

    for (int i = threadIdx.x; i < 292; i += blockDim.x)
        atomicAdd(&ws[i], sAcc[i]);
}

__global__ void gmm_model_build(float* __restrict__ ws, int N) {
    if (threadIdx.x != 0 || blockIdx.x != 0) return;
    float cd_total = 0.f;
    for (int k = 0; k < 4; ++k) {
        const float g = ws[WS_SG + k];
        const float phi = g / (float)N;
        float m[8];
        for (int i = 0; i < 8; ++i) {
            m[i] = ws[WS_SZ + k * 8 + i] / g;
            ws[WS_MU + k * 8 + i] = m[i];
        }
        float c[8][8];
        for (int i = 0; i < 8; ++i)
            for (int j = 0; j < 8; ++j)
                c[i][j] = ws[WS_SZZ + k * 64 + i * 8 + j] / g - m[i] * m[j];
        for (int i = 0; i < 8; ++i) {
            c[i][i] += EPSV;
            cd_total += 1.f / c[i][i];
        }
        // Cholesky c = L L^T
        float L[8][8];
        for (int i = 0; i < 8; ++i)
            for (int j = 0; j < 8; ++j) L[i][j] = 0.f;
        float diagprod = 1.f;
        for (int j = 0; j < 8; ++j) {
            float s = c[j][j];
            for (int p = 0; p < j; ++p) s -= L[j][p] * L[j][p];
            const float lj = sqrtf(s);
            L[j][j] = lj;
            diagprod *= lj;
            const float inv = 1.f / lj;
            for (int i = j + 1; i < 8; ++i) {
                float t = c[i][j];
                for (int p = 0; p < j; ++p) t -= L[i][p] * L[j][p];
                L[i][j] = t * inv;
            }
        }
        // det_cov = prod diag(chol(2*pi*c)) = (2*pi)^(D/2) * prod diag(L)
        const float TWO_PI = 6.2831853071795864f;
        const float tp4 = TWO_PI * TWO_PI * TWO_PI * TWO_PI;
        const float det_cov = tp4 * diagprod;
        ws[WS_COEF + k] = phi / sqrtf(det_cov);
        // inverse via two triangular solves per unit column
        for (int col = 0; col < 8; ++col) {
            float y[8], x[8];
            for (int i = 0; i < 8; ++i) {
                float t = (i == col) ? 1.f : 0.f;
                for (int p = 0; p < i; ++p) t -= L[i][p] * y[p];
                y[i] = t / L[i][i];
            }
            for (int i = 7; i >= 0; --i) {
                float t = y[i];
                for (int p = i + 1; p < 8; ++p) t -= L[p][i] * x[p];
                x[i] = t / L[i][i];
            }
            for (int i = 0; i < 8; ++i)
                ws[WS_P + k * 64 + i * 8 + col] = x[i];
        }
    }
    ws[WS_COVDIAG] = cd_total;
}

__global__ void __launch_bounds__(256)
gmm_energy(const float* __restrict__ z, float* __restrict__ ws, int N) {
    __shared__ float sP[256];
    __shared__ float sMu[32];
    __shared__ float sCoef[4];
    __shared__ float red[256];
    for (int i = threadIdx.x; i < 256; i += blockDim.x) sP[i] = ws[WS_P + i];
    if (threadIdx.x < 32) sMu[threadIdx.x] = ws[WS_MU + threadIdx.x];
    if (threadIdx.x < 4)  sCoef[threadIdx.x] = ws[WS_COEF + threadIdx.x];
    __syncthreads();

    float local = 0.f;
    const int stride = gridDim.x * blockDim.x;
    for (int s = blockIdx.x * blockDim.x + threadIdx.x; s < N; s += stride) {
        const float4 z0 = *(const float4*)(z + s * 8);
        const float4 z1 = *(const float4*)(z + s * 8 + 4);
        const float zz[8] = {z0.x, z0.y, z0.z, z0.w, z1.x, z1.y, z1.z, z1.w};
        float acc = 0.f;
        #pragma unroll
        for (int k = 0; k < 4; ++k) {
            float diff[8];
            #pragma unroll
            for (int i = 0; i < 8; ++i) diff[i] = zz[i] - sMu[k * 8 + i];
            float q = 0.f;
            #pragma unroll
            for (int i = 0; i < 8; ++i) {
                float t = 0.f;
                #pragma unroll
                for (int j = 0; j < 8; ++j)
                    t += sP[k * 64 + i * 8 + j] * diff[j];
                q += diff[i] * t;
            }
            acc += sCoef[k] * expf(-0.5f * q);
        }
        local += -logf(acc + EPSV);
    }

    red[threadIdx.x] = local;
    __syncthreads();
    for (int off = blockDim.x >> 1; off > 0; off >>= 1) {
        if (threadIdx.x < off) red[threadIdx.x] += red[threadIdx.x + off];
        __syncthreads();
    }
    if (threadIdx.x == 0) atomicAdd(&ws[WS_ENERGY], red[0]);
}

__global__ void gmm_finalize(float* __restrict__ out, const float* __restrict__ ws,
                             int N) {
    if (threadIdx.x == 0 && blockIdx.x == 0)
        out[0] = 0.7f * (ws[WS_ENERGY] / (float)N) + 0.3f * ws[WS_COVDIAG];
}

extern "C" void kernel_launch(void* const* d_in, const int* in_sizes, int n_in,
                              void* d_out, int out_size, void* d_ws, size_t ws_size,
                              hipStream_t stream) {
    const float* z     = (const float*)d_in[0];
    const float* gamma = (const float*)d_in[1];
    float* ws  = (float*)d_ws;
    float* out = (float*)d_out;
    const int N = in_sizes[1] / 4;   // gamma is [N,4]

    // zero the accumulator region every launch (ws is poisoned / stale)
    hipMemsetAsync(d_ws, 0, WS_ZERO_FLOATS * sizeof(float), stream);

    gmm_stats_pass<<<BLOCKS1, THREADS1, 0, stream>>>(z, gamma, ws, N);
    gmm_model_build<<<1, 1, 0, stream>>>(ws, N);
    gmm_energy<<<512, 256, 0, stream>>>(z, ws, N);
    gmm_finalize<<<1, 1, 0, stream>>>(out, ws, N);
}